// RbM_19825569038536
// MI455X (gfx1250) — compile-verified
//
#include <hip/hip_runtime.h>
#include <hip/hip_bf16.h>

#define BTOK 16384
#define DDIM 512
#define NEXP 8
#define LOG100 4.605170185988092f
#define MAXT32 (BTOK / 32) // 512 32-token tiles per expert max
#define NRB (BTOK / 8)     // 2048 routing blocks
#define PADENT (2 * BTOK)  // dummy ybuf row index for padded slots

typedef float v2f __attribute__((ext_vector_type(2)));
typedef float v8f __attribute__((ext_vector_type(8)));

__global__ __launch_bounds__(32) void zero_cnt_kernel(int* cnt) {
  if (threadIdx.x < NEXP) cnt[threadIdx.x] = 0;
}

__device__ inline float wave_reduce_add(float v) {
#pragma unroll
  for (int m = 16; m >= 1; m >>= 1) v += __shfl_xor(v, m, 32);
  return v;
}

// One wave per token: normalized routing dists, top-2, softmax gates,
// expert bucket lists (int atomics only), per-block loss partials.
__global__ __launch_bounds__(256) void routing_kernel(
    const float* __restrict__ X, const float* __restrict__ RM,
    int* __restrict__ eidx, float* __restrict__ wgt,
    int* __restrict__ lists, int* __restrict__ cnt,
    float* __restrict__ pvq, float* __restrict__ pattn) {
  __shared__ float rm_s[NEXP * DDIM];
  __shared__ float rinv[NEXP];
  __shared__ int we0[8], we1[8];
  __shared__ float ww0[8], ww1[8], wvq[8];
  const int tid = threadIdx.x;
  for (int i = tid; i < NEXP * DDIM; i += 256) rm_s[i] = RM[i];
  __syncthreads();
  const int wave = tid >> 5, lane = tid & 31;
  {
    float ss = 0.f;
#pragma unroll
    for (int j = 0; j < 16; ++j) {
      float v = rm_s[wave * DDIM + j * 32 + lane];
      ss += v * v;
    }
    ss = wave_reduce_add(ss);
    if (lane == 0) rinv[wave] = 1.0f / fmaxf(sqrtf(ss), 1e-12f);
  }
  __syncthreads();
  const int b = blockIdx.x * 8 + wave;
  float xv[16];
  float ss = 0.f;
#pragma unroll
  for (int j = 0; j < 16; ++j) {
    xv[j] = X[(size_t)b * DDIM + j * 32 + lane];
    ss += xv[j] * xv[j];
  }
  ss = wave_reduce_add(ss);
  const float xinv = 1.0f / fmaxf(sqrtf(ss), 1e-12f);
  float dist[NEXP];
  for (int e = 0; e < NEXP; ++e) {
    float p = 0.f;
#pragma unroll
    for (int j = 0; j < 16; ++j) p += xv[j] * rm_s[e * DDIM + j * 32 + lane];
    p = wave_reduce_add(p);
    dist[e] = p * xinv * rinv[e];
  }
  if (lane == 0) {
    int e0 = 0; float v0 = dist[0];
    for (int e = 1; e < NEXP; ++e) if (dist[e] > v0) { v0 = dist[e]; e0 = e; }
    int e1 = (e0 == 0) ? 1 : 0; float v1 = dist[e1];
    for (int e = 0; e < NEXP; ++e)
      if (e != e0 && dist[e] > v1) { v1 = dist[e]; e1 = e; }
    const float z = expf(v1 - v0);            // softmax over top-2 (m = v0)
    const float w0 = 1.0f / (1.0f + z);
    const float w1 = z / (1.0f + z);
    eidx[2 * b] = e0; eidx[2 * b + 1] = e1;
    wgt[2 * b] = w0;  wgt[2 * b + 1] = w1;
    int p0 = atomicAdd(&cnt[e0], 1);
    lists[e0 * BTOK + p0] = b * 2;            // token*2 + slot
    int p1 = atomicAdd(&cnt[e1], 1);
    lists[e1 * BTOK + p1] = b * 2 + 1;
    we0[wave] = e0; we1[wave] = e1;
    ww0[wave] = w0; ww1[wave] = w1;
    wvq[wave] = -(w0 * v0 + w1 * v1);         // sum_k tv_k * (-top_v_k)
  }
  __syncthreads();
  if (tid < NEXP) {                           // deterministic fixed-order sums
    float a = 0.f;
    for (int w = 0; w < 8; ++w) {
      if (we0[w] == tid) a += ww0[w];
      if (we1[w] == tid) a += ww1[w];
    }
    pattn[blockIdx.x * NEXP + tid] = a;
  }
  if (tid == 0) {
    float s = 0.f;
    for (int w = 0; w < 8; ++w) s += wvq[w];
    pvq[blockIdx.x] = s;
  }
}

// Per-expert GEMM: 32-token tile x 512 outputs, fp32 WMMA 16x16x4.
// A tiles gathered straight into LDS with GLOBAL_LOAD_ASYNC_TO_LDS_B64
// (ASYNCcnt-tracked, no VGPR roundtrip) in the conflict-free float2
// K-pair layout; each B (W) load feeds two WMMAs. Padded slots read
// token 0 and sink their D rows into the write-only ybuf[PADENT] row.
__global__ __launch_bounds__(256) void expert_gemm(
    const float* __restrict__ X, const float* __restrict__ W,
    const int* __restrict__ lists, const int* __restrict__ cnt,
    float* __restrict__ ybuf) {
  const int e = blockIdx.x >> 9;
  const int tile = blockIdx.x & (MAXT32 - 1);
  const int ce = cnt[e];
  if (tile * 32 >= ce) return;
  __shared__ v2f Asa[256 * 16]; // tokens 0..15 : Asa[k2*16+t] = x[t][2k2,2k2+1]
  __shared__ v2f Asb[256 * 16]; // tokens 16..31
  __shared__ int ltok[32];
  const int tid = threadIdx.x;
  if (tid < 32) {
    int idx = tile * 32 + tid;
    ltok[tid] = (idx < ce) ? lists[e * BTOK + idx] : PADENT;
  }
  __syncthreads();
#pragma unroll
  for (int i = 0; i < 32; ++i) { // async-gather 32 tokens x 512 floats -> LDS
    int idx = tid + i * 256;
    int t = idx & 31, k2 = idx >> 5;
    int tok = ltok[t] >> 1;
    if (tok >= BTOK) tok = 0;    // pad row: any valid source, D row is sunk
    const float* gsrc = X + (size_t)tok * DDIM + k2 * 2;
    unsigned lds_dst = (t < 16)
        ? (unsigned)(uintptr_t)&Asa[k2 * 16 + t]
        : (unsigned)(uintptr_t)&Asb[k2 * 16 + (t - 16)];
    asm volatile("global_load_async_to_lds_b64 %0, %1, off"
                 :: "v"(lds_dst), "v"(gsrc)
                 : "memory");
  }
  asm volatile("s_wait_asynccnt 0x0" ::: "memory"); // this wave's copies done
  __syncthreads();                                  // all waves' copies visible
  const int wave = tid >> 5, lane = tid & 31;
  const int half = lane >> 4, lm = lane & 15;
  const float* Wbase = W + (size_t)e * DDIM * DDIM;
  for (int p = 0; p < 4; ++p) {
    const int n = (p << 7) + (wave << 4) + lm;
    const float* Wrow = Wbase + (size_t)n * DDIM + (half << 1);
    v8f c0 = {0.f, 0.f, 0.f, 0.f, 0.f, 0.f, 0.f, 0.f};
    v8f c1 = {0.f, 0.f, 0.f, 0.f, 0.f, 0.f, 0.f, 0.f};
#pragma unroll 4
    for (int s = 0; s < 128; ++s) {
      // A 16x4 fp32 layout: lanes0-15 K=0,1 ; lanes16-31 K=2,3
      v2f a0 = Asa[(2 * s + half) * 16 + lm];
      v2f a1 = Asb[(2 * s + half) * 16 + lm];
      // B 4x16 fp32 layout mirrors A with N across lanes; shared by both tiles
      v2f bb = *(const v2f*)(Wrow + 4 * s);
      c0 = __builtin_amdgcn_wmma_f32_16x16x4_f32(false, a0, false, bb,
                                                 (short)0, c0, false, false);
      c1 = __builtin_amdgcn_wmma_f32_16x16x4_f32(false, a1, false, bb,
                                                 (short)0, c1, false, false);
    }
#pragma unroll
    for (int v = 0; v < 8; ++v) {
      int m = v + (half << 3); // C/D: lanes16-31 hold M = v+8
      ybuf[(size_t)ltok[m] * DDIM + n] = c0[v];       // pad -> dummy row
      ybuf[(size_t)ltok[16 + m] * DDIM + n] = c1[v];  // pad -> dummy row
    }
  }
}

// out = e^min(temp,ln100) * (s0*y0 + s1*y1 + w0*bias[e0] + w1*bias[e1])
__global__ __launch_bounds__(256) void combine_kernel(
    const float* __restrict__ ybuf, const int* __restrict__ eidx,
    const float* __restrict__ wgt, const float* __restrict__ Bias,
    const float* __restrict__ ca, const float* __restrict__ temp,
    float* __restrict__ out_main, float* __restrict__ relu_out) {
  const int b = blockIdx.x;
  const int tid = threadIdx.x;
  const int e0 = eidx[2 * b], e1 = eidx[2 * b + 1];
  const float w0 = wgt[2 * b], w1 = wgt[2 * b + 1];
  const float ts = expf(fminf(temp[0], LOG100));
  const float s0 = w0 * expf(fminf(ca[e0], LOG100));
  const float s1 = w1 * expf(fminf(ca[e1], LOG100));
  for (int o = tid; o < DDIM; o += 256) {
    float y0 = ybuf[(size_t)(2 * b) * DDIM + o];
    float y1 = ybuf[(size_t)(2 * b + 1) * DDIM + o];
    float r = ts * (s0 * y0 + s1 * y1 + w0 * Bias[e0 * DDIM + o] +
                    w1 * Bias[e1 * DDIM + o]);
    out_main[(size_t)b * DDIM + o] = r;
    if (relu_out) relu_out[(size_t)b * DDIM + o] = fmaxf(r, 0.f);
  }
}

__global__ __launch_bounds__(256) void reduce_aux(
    const float* __restrict__ pvq, const float* __restrict__ pattn,
    const float* __restrict__ prev, float* __restrict__ outp) {
  __shared__ float svq[256];
  __shared__ float asum[NEXP];
  const int tid = threadIdx.x;
  float s = 0.f;
  for (int i = tid; i < NRB; i += 256) s += pvq[i];
  svq[tid] = s;
  __syncthreads();
  for (int off = 128; off > 0; off >>= 1) {
    if (tid < off) svq[tid] += svq[tid + off];
    __syncthreads();
  }
  if (tid < NEXP) {
    float a = 0.f;
    for (int i = 0; i < NRB; ++i) a += pattn[i * NEXP + tid];
    asum[tid] = a;
  }
  __syncthreads();
  if (tid == 0) {
    float vq_loss = svq[0] / (float)BTOK;
    float mean = 0.f;
    for (int e = 0; e < NEXP; ++e) mean += asum[e];
    mean /= (float)NEXP;
    float var = 0.f;
    for (int e = 0; e < NEXP; ++e) {
      float d = asum[e] - mean;
      var += d * d;
    }
    var /= (float)(NEXP - 1); // ddof=1
    float lb = var / (mean * mean + 1e-10f);
    float aux = 0.05f * vq_loss + 0.01f * lb;
    if (prev) aux += *prev;
    *outp = aux;
  }
}

extern "C" void kernel_launch(void* const* d_in, const int* in_sizes, int n_in,
                              void* d_out, int out_size, void* d_ws,
                              size_t ws_size, hipStream_t stream) {
  const float* x   = (const float*)d_in[0];
  const float* rm0 = (const float*)d_in[1];
  const float* W0  = (const float*)d_in[2];
  const float* b0  = (const float*)d_in[3];
  const float* t0  = (const float*)d_in[4];
  const float* ca0 = (const float*)d_in[5];
  const float* rm1 = (const float*)d_in[6];
  const float* W1  = (const float*)d_in[7];
  const float* b1  = (const float*)d_in[8];
  const float* t1  = (const float*)d_in[9];
  const float* ca1 = (const float*)d_in[10];
  float* out = (float*)d_out;

  char* ws = (char*)d_ws;
  size_t off = 0;
  // +1 dummy row (index PADENT) as write-only sink for padded tile slots
  float* ybuf = (float*)(ws + off);
  off += ((size_t)(2 * BTOK + 1) * DDIM * 4 + 255) & ~(size_t)255;
  float* h    = (float*)(ws + off); off += (size_t)BTOK * DDIM * 4;
  int*   eidx = (int*)(ws + off);   off += (size_t)BTOK * 2 * 4;
  float* wgt  = (float*)(ws + off); off += (size_t)BTOK * 2 * 4;
  int*   lists= (int*)(ws + off);   off += (size_t)NEXP * BTOK * 4;
  int*   cnt  = (int*)(ws + off);   off += 64;
  float* pvq  = (float*)(ws + off); off += (size_t)NRB * 4;
  float* pattn= (float*)(ws + off); off += (size_t)NRB * NEXP * 4;
  float* aux0 = (float*)(ws + off); off += 64;

  // ---- layer 0 ----
  zero_cnt_kernel<<<1, 32, 0, stream>>>(cnt);
  routing_kernel<<<NRB, 256, 0, stream>>>(x, rm0, eidx, wgt, lists, cnt, pvq, pattn);
  expert_gemm<<<NEXP * MAXT32, 256, 0, stream>>>(x, W0, lists, cnt, ybuf);
  combine_kernel<<<BTOK, 256, 0, stream>>>(ybuf, eidx, wgt, b0, ca0, t0,
                                           out /*h_emb*/, h /*relu*/);
  reduce_aux<<<1, 256, 0, stream>>>(pvq, pattn, nullptr, aux0);

  // ---- layer 1 ----
  zero_cnt_kernel<<<1, 32, 0, stream>>>(cnt);
  routing_kernel<<<NRB, 256, 0, stream>>>(h, rm1, eidx, wgt, lists, cnt, pvq, pattn);
  expert_gemm<<<NEXP * MAXT32, 256, 0, stream>>>(h, W1, lists, cnt, ybuf);
  combine_kernel<<<BTOK, 256, 0, stream>>>(ybuf, eidx, wgt, b1, ca1, t1,
                                           out + (size_t)BTOK * DDIM, nullptr);
  reduce_aux<<<1, 256, 0, stream>>>(pvq, pattn, aux0,
                                    out + (size_t)2 * BTOK * DDIM);
}